// DynamicLoss_48859547959508
// MI455X (gfx1250) — compile-verified
//
#include <hip/hip_runtime.h>
#include <hip/hip_bf16.h>
#include <math.h>

// Problem sizes (fixed by the reference)
#define NB   16      // batches
#define NP   16384   // points
#define NOBJ 32      // objects (K)
#define NF   16      // reduction features: [1, p1(3), p2(3), p1_i*p2_j(9)]
#define CH1  64      // phase-1 chunks of N per batch (1 wave each)
#define CH3  32      // phase-3 chunks of N per batch (1 block each)

typedef float v2f __attribute__((ext_vector_type(2)));
typedef float v8f __attribute__((ext_vector_type(8)));

__device__ __forceinline__ float sel4(int i, float a, float b, float c, float d) {
  float r = a;
  r = (i == 1) ? b : r;
  r = (i == 2) ? c : r;
  r = (i == 3) ? d : r;
  return r;
}

__device__ __forceinline__ bool finitef(float x) {
  return (x == x) && (fabsf(x) <= 3.0e38f);
}

// ---------------------------------------------------------------------------
// Phase 1: per (batch, chunk) wave computes partial (32 obj x 16 feat) sums
// via V_WMMA_F32_16X16X4_F32:  D(16x16) += A(16objs x 4pts) * B(4pts x 16feat)
// ---------------------------------------------------------------------------
__global__ __launch_bounds__(32) void k_phase1(const float* __restrict__ pc,
                                               const float* __restrict__ mask,
                                               const float* __restrict__ flow,
                                               float* __restrict__ part) {
  const int b    = blockIdx.x;
  const int ch   = blockIdx.y;
  const int lane = threadIdx.x;
  const int m16  = lane & 15;
  const int hi   = lane >> 4;

  // Feature f = m16 decomposes into product a1[fi] * a2[fj] over {1,x,y,z}
  const int f = m16;
  int fi, fj;
  if (f == 0)      { fi = 0;              fj = 0; }
  else if (f <= 3) { fi = f;              fj = 0; }
  else if (f <= 6) { fi = 0;              fj = f - 3; }
  else             { fi = 1 + (f - 7) / 3; fj = 1 + (f - 7) % 3; }

  const float* __restrict__ pcB = pc   + (size_t)b * NP * 3;
  const float* __restrict__ flB = flow + (size_t)b * NP * 3;
  const float* __restrict__ mkB = mask + (size_t)b * NP * NOBJ;

  v8f c0 = {0.f,0.f,0.f,0.f,0.f,0.f,0.f,0.f};   // objects 0..15
  v8f c1 = c0;                                   // objects 16..31

  const int ptsPerWave = NP / CH1;   // 256
  const int steps      = ptsPerWave / 4;
  const int pbase      = ch * ptsPerWave;

  for (int s = 0; s < steps; ++s) {
    const int pa = pbase + s * 4 + 2 * hi;   // this lane's first point of its K-pair

    // Coordinates for points pa and pa+1 (broadcast across the 16 lanes of a half)
    float x1a = pcB[pa*3+0], y1a = pcB[pa*3+1], z1a = pcB[pa*3+2];
    float x1b = pcB[pa*3+3], y1b = pcB[pa*3+4], z1b = pcB[pa*3+5];
    float x2a = x1a + flB[pa*3+0], y2a = y1a + flB[pa*3+1], z2a = z1a + flB[pa*3+2];
    float x2b = x1b + flB[pa*3+3], y2b = y1b + flB[pa*3+4], z2b = z1b + flB[pa*3+5];

    // B operand: feature m16 of points pa, pa+1
    float fa = sel4(fi, 1.f, x1a, y1a, z1a) * sel4(fj, 1.f, x2a, y2a, z2a);
    float fb = sel4(fi, 1.f, x1b, y1b, z1b) * sel4(fj, 1.f, x2b, y2b, z2b);
    v2f bv = {fa, fb};

    // A operands: mask^T tiles. A[m][k] = mask[point pa..][obj m16 (+16)]
    const float* ma = mkB + (size_t)pa * NOBJ;
    v2f a0 = { ma[m16],      ma[NOBJ + m16] };
    v2f a1 = { ma[16 + m16], ma[NOBJ + 16 + m16] };

    c0 = __builtin_amdgcn_wmma_f32_16x16x4_f32(false, a0, false, bv, (short)0, c0, false, false);
    c1 = __builtin_amdgcn_wmma_f32_16x16x4_f32(false, a1, false, bv, (short)0, c1, false, false);
  }

  // Write this wave's partial tile (no atomics: disjoint slots, deterministic)
  float* __restrict__ dst = part + (size_t)(b * CH1 + ch) * (NOBJ * NF);
  #pragma unroll
  for (int r = 0; r < 8; ++r) {
    dst[(r + 8*hi)      * NF + m16] = c0[r];
    dst[(16 + r + 8*hi) * NF + m16] = c1[r];
  }
}

// ---------------------------------------------------------------------------
// Reduce phase-1 partials over chunks (fixed order => deterministic)
// acc[(b*NOBJ+k)*NF + f] = sum_ch part[b][ch][k][f]
// ---------------------------------------------------------------------------
__global__ __launch_bounds__(256) void k_reduce(const float* __restrict__ part,
                                                float* __restrict__ acc) {
  const int idx = blockIdx.x * 256 + threadIdx.x;      // 0..8191
  const int b   = idx >> 9;                            // / (NOBJ*NF)
  const int rem = idx & 511;
  const float* __restrict__ p = part + (size_t)b * CH1 * (NOBJ * NF) + rem;
  float sum = 0.f;
  for (int ch = 0; ch < CH1; ++ch) sum += p[(size_t)ch * (NOBJ * NF)];
  acc[idx] = sum;
}

// ---------------------------------------------------------------------------
// Phase 2: 512 weighted Kabsch fits. One thread per (b,k).
// ---------------------------------------------------------------------------
__device__ __forceinline__ void jrot(float A[3][3], float V[3][3], int p, int q) {
  float apq = A[p][q];
  if (fabsf(apq) < 1e-30f) return;
  float tau = (A[q][q] - A[p][p]) / (2.0f * apq);
  float tt  = copysignf(1.0f, tau) / (fabsf(tau) + sqrtf(1.0f + tau * tau));
  float c   = 1.0f / sqrtf(1.0f + tt * tt);
  float sn  = tt * c;
  #pragma unroll
  for (int k = 0; k < 3; ++k) {
    float akp = A[k][p], akq = A[k][q];
    A[k][p] = c * akp - sn * akq;
    A[k][q] = sn * akp + c * akq;
  }
  #pragma unroll
  for (int k = 0; k < 3; ++k) {
    float apk = A[p][k], aqk = A[q][k];
    A[p][k] = c * apk - sn * aqk;
    A[q][k] = sn * apk + c * aqk;
  }
  #pragma unroll
  for (int k = 0; k < 3; ++k) {
    float vkp = V[k][p], vkq = V[k][q];
    V[k][p] = c * vkp - sn * vkq;
    V[k][q] = sn * vkp + c * vkq;
  }
}

__device__ __forceinline__ float det3(const float M[3][3]) {
  return M[0][0] * (M[1][1]*M[2][2] - M[1][2]*M[2][1])
       - M[0][1] * (M[1][0]*M[2][2] - M[1][2]*M[2][0])
       + M[0][2] * (M[1][0]*M[2][1] - M[1][1]*M[2][0]);
}

__device__ __forceinline__ void swap_col(float V[3][3], float* lam, int a, int b) {
  float tl = lam[a]; lam[a] = lam[b]; lam[b] = tl;
  #pragma unroll
  for (int k = 0; k < 3; ++k) { float tv = V[k][a]; V[k][a] = V[k][b]; V[k][b] = tv; }
}

__global__ __launch_bounds__(256) void k_fit(const float* __restrict__ acc,
                                             float* __restrict__ rt) {
  const int idx = blockIdx.x * 256 + threadIdx.x;
  if (idx >= NB * NOBJ) return;
  const float* __restrict__ s = acc + (size_t)idx * NF;

  float sv[NF];
  #pragma unroll
  for (int i = 0; i < NF; ++i) sv[i] = s[i];

  const float w = sv[0];
  bool ok = finitef(w) && (fabsf(w) > 1e-12f);
  const float invw = 1.0f / w;

  float mu1[3], mu2[3];
  #pragma unroll
  for (int i = 0; i < 3; ++i) { mu1[i] = sv[1 + i] * invw; mu2[i] = sv[4 + i] * invw; }

  float Sm[3][3];
  #pragma unroll
  for (int i = 0; i < 3; ++i)
    #pragma unroll
    for (int j = 0; j < 3; ++j) {
      Sm[i][j] = sv[7 + 3*i + j] - w * mu1[i] * mu2[j];
      ok = ok && finitef(Sm[i][j]);
    }

  // A = S^T S (symmetric), Jacobi eigensolve -> V, eigenvalues
  float Am[3][3];
  #pragma unroll
  for (int i = 0; i < 3; ++i)
    #pragma unroll
    for (int j = 0; j < 3; ++j)
      Am[i][j] = Sm[0][i]*Sm[0][j] + Sm[1][i]*Sm[1][j] + Sm[2][i]*Sm[2][j];

  float Vm[3][3] = {{1.f,0.f,0.f},{0.f,1.f,0.f},{0.f,0.f,1.f}};
  for (int sw = 0; sw < 8; ++sw) {
    jrot(Am, Vm, 0, 1);
    jrot(Am, Vm, 0, 2);
    jrot(Am, Vm, 1, 2);
  }
  float lam[3] = {Am[0][0], Am[1][1], Am[2][2]};
  if (lam[0] < lam[1]) swap_col(Vm, lam, 0, 1);
  if (lam[0] < lam[2]) swap_col(Vm, lam, 0, 2);
  if (lam[1] < lam[2]) swap_col(Vm, lam, 1, 2);

  // U columns: normalize S * v_i ; sigma_i = |S v_i|
  float Um[3][3], sig[3];
  #pragma unroll
  for (int c = 0; c < 3; ++c) {
    float ux = Sm[0][0]*Vm[0][c] + Sm[0][1]*Vm[1][c] + Sm[0][2]*Vm[2][c];
    float uy = Sm[1][0]*Vm[0][c] + Sm[1][1]*Vm[1][c] + Sm[1][2]*Vm[2][c];
    float uz = Sm[2][0]*Vm[0][c] + Sm[2][1]*Vm[1][c] + Sm[2][2]*Vm[2][c];
    float n  = sqrtf(ux*ux + uy*uy + uz*uz);
    sig[c]   = n;
    float inv = 1.0f / fmaxf(n, 1e-30f);
    Um[0][c] = ux * inv; Um[1][c] = uy * inv; Um[2][c] = uz * inv;
  }
  if (!finitef(sig[2]) || sig[2] < 1e-6f * fmaxf(sig[0], 1e-30f)) {
    // rebuild smallest left-singular direction as u0 x u1
    float cx = Um[1][0]*Um[2][1] - Um[2][0]*Um[1][1];
    float cy = Um[2][0]*Um[0][1] - Um[0][0]*Um[2][1];
    float cz = Um[0][0]*Um[1][1] - Um[1][0]*Um[0][1];
    float inv = 1.0f / fmaxf(sqrtf(cx*cx + cy*cy + cz*cz), 1e-30f);
    Um[0][2] = cx * inv; Um[1][2] = cy * inv; Um[2][2] = cz * inv;
  }

  const float d = (det3(Vm) * det3(Um)) < 0.f ? -1.f : 1.f;

  float R[3][3], t[3];
  #pragma unroll
  for (int i = 0; i < 3; ++i)
    #pragma unroll
    for (int j = 0; j < 3; ++j)
      R[i][j] = Vm[i][0]*Um[j][0] + Vm[i][1]*Um[j][1] + d * Vm[i][2]*Um[j][2];
  #pragma unroll
  for (int i = 0; i < 3; ++i) {
    t[i] = mu2[i] - (R[i][0]*mu1[0] + R[i][1]*mu1[1] + R[i][2]*mu1[2]);
    ok = ok && finitef(t[i]);
  }
  #pragma unroll
  for (int i = 0; i < 3; ++i)
    #pragma unroll
    for (int j = 0; j < 3; ++j)
      ok = ok && finitef(R[i][j]);

  if (!ok) {
    #pragma unroll
    for (int i = 0; i < 3; ++i) {
      #pragma unroll
      for (int j = 0; j < 3; ++j) R[i][j] = (i == j) ? 1.f : 0.f;
      t[i] = 0.f;
    }
  }

  float* __restrict__ o = rt + (size_t)idx * 12;
  #pragma unroll
  for (int i = 0; i < 3; ++i)
    #pragma unroll
    for (int j = 0; j < 3; ++j)
      o[3*i + j] = R[i][j];
  o[9] = t[0]; o[10] = t[1]; o[11] = t[2];
}

// ---------------------------------------------------------------------------
// Phase 3: blend GEMM mask(Nx32) x RT(32x12->16) via WMMA, then per-point L2.
// Each block: 8 waves x 4 tiles of 16 points = 512 points.
// ---------------------------------------------------------------------------
__global__ __launch_bounds__(256) void k_phase3(const float* __restrict__ pc,
                                                const float* __restrict__ mask,
                                                const float* __restrict__ flow,
                                                const float* __restrict__ rt,
                                                float* __restrict__ lpart) {
  __shared__ float lds[8][16][16];
  __shared__ float red[256];

  const int b   = blockIdx.x;
  const int ch  = blockIdx.y;
  const int tid = threadIdx.x;
  const int w   = tid >> 5;
  const int lane = tid & 31;
  const int m16 = lane & 15;
  const int hi  = lane >> 4;

  // Preload B operands (loop-invariant): B[k][col] = RT[b][kk*4+k][col], cols>=12 are 0
  const float* __restrict__ rtB = rt + (size_t)b * NOBJ * 12;
  v2f bv[8];
  #pragma unroll
  for (int kk = 0; kk < 8; ++kk) {
    const int k0 = kk * 4 + 2 * hi;
    float bx = (m16 < 12) ? rtB[k0 * 12 + m16]       : 0.f;
    float by = (m16 < 12) ? rtB[(k0 + 1) * 12 + m16] : 0.f;
    bv[kk] = {bx, by};
  }

  const float* __restrict__ mkB = mask + (size_t)b * NP * NOBJ;
  const float* __restrict__ pcB = pc   + (size_t)b * NP * 3;
  const float* __restrict__ flB = flow + (size_t)b * NP * 3;

  float lsum = 0.f;
  const int pwave = ch * (NP / CH3) + w * 64;

  for (int tile = 0; tile < 4; ++tile) {
    const int p0 = pwave + tile * 16;

    v8f c = {0.f,0.f,0.f,0.f,0.f,0.f,0.f,0.f};
    #pragma unroll
    for (int kk = 0; kk < 8; ++kk) {
      const float* mrow = mkB + (size_t)(p0 + m16) * NOBJ + kk * 4 + 2 * hi;
      v2f a = { mrow[0], mrow[1] };
      c = __builtin_amdgcn_wmma_f32_16x16x4_f32(false, a, false, bv[kk], (short)0, c, false, false);
    }

    #pragma unroll
    for (int r = 0; r < 8; ++r) lds[w][r + 8*hi][m16] = c[r];
    __syncthreads();

    if (hi == 0) {
      const int p = p0 + m16;
      const float* row = lds[w][m16];
      float x1 = pcB[p*3+0], y1 = pcB[p*3+1], z1 = pcB[p*3+2];
      float x2 = x1 + flB[p*3+0], y2 = y1 + flB[p*3+1], z2 = z1 + flB[p*3+2];
      float px = row[0]*x1 + row[1]*y1 + row[2]*z1 + row[9];
      float py = row[3]*x1 + row[4]*y1 + row[5]*z1 + row[10];
      float pz = row[6]*x1 + row[7]*y1 + row[8]*z1 + row[11];
      float dx = px - x2, dy = py - y2, dz = pz - z2;
      lsum += sqrtf(dx*dx + dy*dy + dz*dz);
    }
    __syncthreads();
  }

  // Block reduction (fixed pairing => deterministic)
  red[tid] = lsum;
  __syncthreads();
  for (int s2 = 128; s2 > 0; s2 >>= 1) {
    if (tid < s2) red[tid] += red[tid + s2];
    __syncthreads();
  }
  if (tid == 0) lpart[b * CH3 + ch] = red[0];
}

// ---------------------------------------------------------------------------
// Final: sum 512 block partials (fixed tree) and scale by 1/(B*N)
// ---------------------------------------------------------------------------
__global__ __launch_bounds__(256) void k_final(const float* __restrict__ lpart,
                                               float* __restrict__ out) {
  __shared__ float red[256];
  const int t = threadIdx.x;
  red[t] = lpart[t] + lpart[t + 256];
  __syncthreads();
  for (int s2 = 128; s2 > 0; s2 >>= 1) {
    if (t < s2) red[t] += red[t + s2];
    __syncthreads();
  }
  if (t == 0) out[0] = red[0] * (1.0f / (float)(NB * NP));
}

extern "C" void kernel_launch(void* const* d_in, const int* in_sizes, int n_in,
                              void* d_out, int out_size, void* d_ws, size_t ws_size,
                              hipStream_t stream) {
  const float* pc   = (const float*)d_in[0];
  const float* mask = (const float*)d_in[1];
  const float* flow = (const float*)d_in[2];
  float* out = (float*)d_out;

  // Workspace layout (floats): part1 | acc | rt | lossPart
  float* part1 = (float*)d_ws;                               // NB*CH1*NOBJ*NF = 524288
  float* acc   = part1 + (size_t)NB * CH1 * NOBJ * NF;       // 8192
  float* rtbuf = acc   + (size_t)NB * NOBJ * NF;             // 6144
  float* lpart = rtbuf + (size_t)NB * NOBJ * 12;             // 512

  k_phase1<<<dim3(NB, CH1), 32, 0, stream>>>(pc, mask, flow, part1);
  k_reduce<<<(NB * NOBJ * NF) / 256, 256, 0, stream>>>(part1, acc);
  k_fit<<<(NB * NOBJ + 255) / 256, 256, 0, stream>>>(acc, rtbuf);
  k_phase3<<<dim3(NB, CH3), 256, 0, stream>>>(pc, mask, flow, rtbuf, lpart);
  k_final<<<1, 256, 0, stream>>>(lpart, out);
}